// SMPL_query_27487790695221
// MI455X (gfx1250) — compile-verified
//
#include <hip/hip_runtime.h>
#include <hip/hip_bf16.h>

typedef __attribute__((ext_vector_type(2))) float v2f;
typedef __attribute__((ext_vector_type(8))) float v8f;

#define Bn 2
#define Sn 1024
#define Vn 6890
#define Fn 13776
#define NTILE 861          // Fn / 16 exactly
#define NSLICE 8           // face slices across gridDim.y
#define NWAVE 8            // waves per block (256 threads, wave32)

// divisor is pre-sanitized (never 0) -> single v_rcp_f32 (TRANS pipe) + mul.
// Always called UNCONDITIONALLY so the compiler never has to branch around it.
__device__ __forceinline__ float safediv(float n, float d) {
    return n * __builtin_amdgcn_rcpf((d == 0.0f) ? 1.0f : d);
}

// ---------------------------------------------------------------------------
// Scan kernel: one wave = 16 query points (M tile). Loop over 16-face N tiles.
// Three V_WMMA_F32_16X16X4_F32 per tile compute ab.p, ac.p, a.p for all
// 16x16 (point, face) pairs; branchless VALU does the Voronoi-region
// classification and the running argmin of dist^2.
// ---------------------------------------------------------------------------
__global__ __launch_bounds__(256) void cp_scan_kernel(
    const float* __restrict__ coords,
    const float* __restrict__ smpl_V,
    const int*   __restrict__ smpl_F,
    float* __restrict__ pd2, int* __restrict__ pfid,
    float* __restrict__ pv,  float* __restrict__ pw)
{
    __shared__ float s_d2[NWAVE][16];
    __shared__ int   s_f [NWAVE][16];
    __shared__ float s_v [NWAVE][16];
    __shared__ float s_w [NWAVE][16];

    const int tid   = threadIdx.x;
    const int lane  = tid & 31;
    const int wave  = tid >> 5;
    const int col   = lane & 15;      // N column (face within tile) / A row
    const bool hi   = lane >= 16;     // upper half-wave: K=2,3 and C rows 8..15
    const int mbase = hi ? 8 : 0;

    const int blk   = blockIdx.x;     // 0..127 : (batch, point tile)
    const int slice = blockIdx.y;     // 0..NSLICE-1 : face slice
    const int b     = blk >> 6;
    const int s0    = (blk & 63) << 4;

    const float* __restrict__ Pb = coords + ((size_t)b * Sn + s0) * 3;

    // A operand (16x4 f32): row = point, K = {x, y, z, 0}
    const float px = Pb[3*col+0], py = Pb[3*col+1], pz = Pb[3*col+2];
    v2f aP;
    aP.x = hi ? pz : px;
    aP.y = hi ? 0.0f : py;

    // |p|^2 for the 8 C rows this lane owns
    float pp[8];
#pragma unroll
    for (int r = 0; r < 8; ++r) {
        const float qx = Pb[3*(mbase+r)+0];
        const float qy = Pb[3*(mbase+r)+1];
        const float qz = Pb[3*(mbase+r)+2];
        pp[r] = qx*qx + qy*qy + qz*qz;
    }

    float bd2[8]; int bf[8]; float bv[8], bw[8];
#pragma unroll
    for (int r = 0; r < 8; ++r) { bd2[r] = 3.4e38f; bf[r] = 0x7fffffff; bv[r] = 0.f; bw[r] = 0.f; }

    const float* __restrict__ Vb = smpl_V + (size_t)b * Vn * 3;
    const v8f czero = {0.f,0.f,0.f,0.f,0.f,0.f,0.f,0.f};

    const int wslice = slice * NWAVE + wave;     // 0..63

    for (int t = wslice; t < NTILE; t += NSLICE * NWAVE) {
        const int f  = (t << 4) + col;
        const int i0 = smpl_F[3*f+0];
        const int i1 = smpl_F[3*f+1];
        const int i2 = smpl_F[3*f+2];
        const float ax = Vb[3*i0+0], ay = Vb[3*i0+1], az = Vb[3*i0+2];
        const float bx = Vb[3*i1+0], by = Vb[3*i1+1], bz = Vb[3*i1+2];
        const float cx = Vb[3*i2+0], cy = Vb[3*i2+1], cz = Vb[3*i2+2];
        const float abx = bx-ax, aby = by-ay, abz = bz-az;
        const float acx = cx-ax, acy = cy-ay, acz = cz-az;
        // per-face scalar constants (shared by the 8 rows of this lane)
        const float ab_a = abx*ax + aby*ay + abz*az;
        const float ac_a = acx*ax + acy*ay + acz*az;
        const float ab_b = abx*bx + aby*by + abz*bz;
        const float ac_b = acx*bx + acy*by + acz*bz;
        const float ab_c = abx*cx + aby*cy + abz*cz;
        const float ac_c = acx*cx + acy*cy + acz*cz;
        const float a_a  = ax*ax + ay*ay + az*az;
        const float ab2  = abx*abx + aby*aby + abz*abz;
        const float ac2  = acx*acx + acy*acy + acz*acz;
        const float abac = abx*acx + aby*acy + abz*acz;

        // B operands (4x16 f32): column = face, K = edge components
        v2f bAB, bAC, bA;
        bAB.x = hi ? abz : abx;  bAB.y = hi ? 0.f : aby;
        bAC.x = hi ? acz : acx;  bAC.y = hi ? 0.f : acy;
        bA.x  = hi ? az  : ax;   bA.y  = hi ? 0.f : ay;

        v8f tAB = __builtin_amdgcn_wmma_f32_16x16x4_f32(false, aP, false, bAB, (short)0, czero, false, false);
        v8f tAC = __builtin_amdgcn_wmma_f32_16x16x4_f32(false, aP, false, bAC, (short)0, czero, false, false);
        v8f tA  = __builtin_amdgcn_wmma_f32_16x16x4_f32(false, aP, false, bA,  (short)0, czero, false, false);

#pragma unroll
        for (int r = 0; r < 8; ++r) {
            const float abp = tAB[r], acp = tAC[r], ap = tA[r];
            const float d1 = abp - ab_a;
            const float d2 = acp - ac_a;
            const float d3 = abp - ab_b;
            const float d4 = acp - ac_b;
            const float d5 = abp - ab_c;
            const float d6 = acp - ac_c;
            const float va = d3*d6 - d5*d4;
            const float vb = d5*d2 - d1*d6;
            const float vc = d1*d4 - d3*d2;
            const float denom = va + vb + vc;
            const float e43 = d4 - d3, e56 = d5 - d6;

            // ALL divisions hoisted & unconditional -> no EXEC branching
            const float v0 = safediv(vb, denom);
            const float w0 = safediv(vc, denom);
            const float t1 = safediv(e43, e43 + e56);
            const float w2 = safediv(d2, d2 - d6);
            const float v3 = safediv(d1, d1 - d3);

            float v = v0, w = w0;
            bool m = (va <= 0.f) && (e43 >= 0.f) && (e56 >= 0.f);
            v = m ? (1.f - t1) : v;  w = m ? t1 : w;
            m = (vb <= 0.f) && (d2 >= 0.f) && (d6 <= 0.f);
            v = m ? 0.f : v;  w = m ? w2 : w;
            m = (vc <= 0.f) && (d1 >= 0.f) && (d3 <= 0.f);
            v = m ? v3 : v;   w = m ? 0.f : w;
            m = (d6 >= 0.f) && (d5 <= d6);
            v = m ? 0.f : v;  w = m ? 1.f : w;
            m = (d3 >= 0.f) && (d4 <= d3);
            v = m ? 1.f : v;  w = m ? 0.f : w;
            m = (d1 <= 0.f) && (d2 <= 0.f);
            v = m ? 0.f : v;  w = m ? 0.f : w;

            const float papa = pp[r] - 2.f*ap + a_a;
            float dist2 = papa - 2.f*v*d1 - 2.f*w*d2
                        + v*v*ab2 + 2.f*v*w*abac + w*w*ac2;
            dist2 = fmaxf(dist2, 0.f);

            // faces are visited in increasing fid order per lane, so strict <
            // keeps the first (lowest-fid) minimum, matching jnp.argmin.
            // Explicit select form -> v_cndmask, no branch.
            const bool upd = dist2 < bd2[r];
            bd2[r] = upd ? dist2 : bd2[r];
            bf[r]  = upd ? f     : bf[r];
            bv[r]  = upd ? v     : bv[r];
            bw[r]  = upd ? w     : bw[r];
        }
    }

    // Butterfly min over the 16 face columns (stays inside each 16-lane half).
    // Tie-break toward lower fid to match argmin-first semantics.
#pragma unroll
    for (int r = 0; r < 8; ++r) {
#pragma unroll
        for (int mask = 1; mask < 16; mask <<= 1) {
            const float od2 = __shfl_xor(bd2[r], mask, 32);
            const int   of  = __shfl_xor(bf[r],  mask, 32);
            const float ov  = __shfl_xor(bv[r],  mask, 32);
            const float ow  = __shfl_xor(bw[r],  mask, 32);
            const bool upd = (od2 < bd2[r]) || (od2 == bd2[r] && of < bf[r]);
            bd2[r] = upd ? od2 : bd2[r];
            bf[r]  = upd ? of  : bf[r];
            bv[r]  = upd ? ov  : bv[r];
            bw[r]  = upd ? ow  : bw[r];
        }
    }

    if (col == 0) {
#pragma unroll
        for (int r = 0; r < 8; ++r) {
            s_d2[wave][mbase+r] = bd2[r];
            s_f [wave][mbase+r] = bf[r];
            s_v [wave][mbase+r] = bv[r];
            s_w [wave][mbase+r] = bw[r];
        }
    }
    __syncthreads();

    if (tid < 16) {
        float d2m = s_d2[0][tid]; int fm = s_f[0][tid];
        float vm = s_v[0][tid],  wm = s_w[0][tid];
#pragma unroll
        for (int q = 1; q < NWAVE; ++q) {
            const float d = s_d2[q][tid];
            const int   fq = s_f[q][tid];
            const bool upd = (d < d2m) || (d == d2m && fq < fm);
            d2m = upd ? d : d2m;
            fm  = upd ? fq : fm;
            vm  = upd ? s_v[q][tid] : vm;
            wm  = upd ? s_w[q][tid] : wm;
        }
        const size_t po = (size_t)slice * (Bn*Sn) + (size_t)b * Sn + s0 + tid;
        pd2[po] = d2m; pfid[po] = fm; pv[po] = vm; pw[po] = wm;
    }
}

// ---------------------------------------------------------------------------
// Epilogue: merge the NSLICE partial argmins and build all four outputs.
// Layout of d_out (floats): out_coord[0..6143] | sdf[6144..8191]
//                           | normal[8192..14335] | z[14336..16383]
// ---------------------------------------------------------------------------
__global__ __launch_bounds__(256) void cp_out_kernel(
    const float* __restrict__ coords,
    const float* __restrict__ smpl_V,
    const float* __restrict__ can_V,
    const int*   __restrict__ smpl_F,
    const float* __restrict__ pd2, const int* __restrict__ pfid,
    const float* __restrict__ pv,  const float* __restrict__ pw,
    float* __restrict__ out)
{
    const int idx = blockIdx.x * blockDim.x + threadIdx.x;
    if (idx >= Bn*Sn) return;
    const int b = idx >> 10;   // Sn == 1024

    float d2m = pd2[idx]; int fm = pfid[idx]; float v = pv[idx], w = pw[idx];
#pragma unroll
    for (int sl = 1; sl < NSLICE; ++sl) {
        const int o = sl*(Bn*Sn) + idx;
        const float d = pd2[o];
        const int   fq = pfid[o];
        const bool upd = (d < d2m) || (d == d2m && fq < fm);
        d2m = upd ? d : d2m;
        fm  = upd ? fq : fm;
        v   = upd ? pv[o] : v;
        w   = upd ? pw[o] : w;
    }

    const float px = coords[3*idx+0], py = coords[3*idx+1], pz = coords[3*idx+2];
    const float* __restrict__ Vb = smpl_V + (size_t)b * Vn * 3;
    const int i0 = smpl_F[3*fm+0], i1 = smpl_F[3*fm+1], i2 = smpl_F[3*fm+2];
    const float ax = Vb[3*i0+0], ay = Vb[3*i0+1], az = Vb[3*i0+2];
    const float bx = Vb[3*i1+0], by = Vb[3*i1+1], bz = Vb[3*i1+2];
    const float cx = Vb[3*i2+0], cy = Vb[3*i2+1], cz = Vb[3*i2+2];

    const float hx = ax + v*(bx-ax) + w*(cx-ax);
    const float hy = ay + v*(by-ay) + w*(cy-ay);
    const float hz = az + v*(bz-az) + w*(cz-az);
    const float dx = hx-px, dy = hy-py, dz = hz-pz;
    const float nrm = fmaxf(sqrtf(dx*dx + dy*dy + dz*dz), 1e-6f);
    const float rn  = 1.0f / nrm;
    const float w0 = 1.f - v - w;

    const float ox = w0*can_V[3*i0+0] + v*can_V[3*i1+0] + w*can_V[3*i2+0];
    const float oy = w0*can_V[3*i0+1] + v*can_V[3*i1+1] + w*can_V[3*i2+1];
    const float oz = w0*can_V[3*i0+2] + v*can_V[3*i1+2] + w*can_V[3*i2+2];

    out[3*idx+0] = ox; out[3*idx+1] = oy; out[3*idx+2] = oz;   // out_coord
    out[6144 + idx] = sqrtf(d2m);                              // sdf
    out[8192 + 3*idx+0] = dx*rn;                               // normal
    out[8192 + 3*idx+1] = dy*rn;
    out[8192 + 3*idx+2] = dz*rn;
    out[14336 + idx] = pz;                                     // z
}

// ---------------------------------------------------------------------------
extern "C" void kernel_launch(void* const* d_in, const int* in_sizes, int n_in,
                              void* d_out, int out_size, void* d_ws, size_t ws_size,
                              hipStream_t stream) {
    const float* coords = (const float*)d_in[0];
    const float* smpl_V = (const float*)d_in[1];
    const float* can_V  = (const float*)d_in[2];
    const int*   smpl_F = (const int*)d_in[3];
    float* out = (float*)d_out;

    // workspace: NSLICE partial argmin arrays (d2, fid, v, w) -> 256 KB total
    const size_t N = (size_t)NSLICE * Bn * Sn;
    float* pd2  = (float*)d_ws;
    int*   pfid = (int*)  ((char*)d_ws + N*4);
    float* pv   = (float*)((char*)d_ws + N*8);
    float* pw   = (float*)((char*)d_ws + N*12);

    dim3 grid(Bn * (Sn/16), NSLICE);
    cp_scan_kernel<<<grid, 256, 0, stream>>>(coords, smpl_V, smpl_F, pd2, pfid, pv, pw);
    cp_out_kernel<<<(Bn*Sn + 255)/256, 256, 0, stream>>>(coords, smpl_V, can_V, smpl_F,
                                                         pd2, pfid, pv, pw, out);
}